// OccupancyGrid_11587821765183
// MI455X (gfx1250) — compile-verified
//
#include <hip/hip_runtime.h>

#define RES      128
#define NCELLS   (RES * RES * RES)       // 2,097,152
#define NWORDS   (NCELLS / 32)           // 65,536 u32 = 256 KB bitmask
#define NCHUNK   (NWORDS / 4)            // 16,384 uint4 chunks
#define THRESH   0.01f

#if defined(__gfx1250__) && __has_builtin(__builtin_amdgcn_global_load_async_to_lds_b128)
#define HAVE_ASYNC_LDS 1
#else
#define HAVE_ASYNC_LDS 0
#endif

typedef int v4i __attribute__((vector_size(16)));   // matches builtin param type

__device__ __forceinline__ int clamp127(int v) {
    return v < 0 ? 0 : (v > 127 ? 127 : v);
}

// K1: copy grid -> updated (d_out), pack occupancy bits -> gmask (d_ws).
// One thread per 32 consecutive cells (= one mask word, z fastest dim).
__global__ void og_pack_copy(const float* __restrict__ grid,
                             float* __restrict__ updated,
                             unsigned* __restrict__ gmask) {
    int t = blockIdx.x * blockDim.x + threadIdx.x;        // 0..65535
    const float4* g4 = (const float4*)grid;
    float4*       u4 = (float4*)updated;
    unsigned w = 0u;
#pragma unroll
    for (int j = 0; j < 8; ++j) {
        float4 v = g4[t * 8 + j];
        u4[t * 8 + j] = v;
        if (v.x > 0.0f) w |= 1u << (j * 4 + 0);
        if (v.y > 0.0f) w |= 1u << (j * 4 + 1);
        if (v.z > 0.0f) w |= 1u << (j * 4 + 2);
        if (v.w > 0.0f) w |= 1u << (j * 4 + 3);
    }
    gmask[t] = w;
}

// K2: scatter-max. All writers store identical 1.0f -> plain store is race-free.
// Bitmask updated with global_atomic_or.
__global__ void og_scatter(const float* __restrict__ coords,
                           const float* __restrict__ dens,
                           float* __restrict__ updated,
                           unsigned* __restrict__ gmask, int n) {
    int i = blockIdx.x * blockDim.x + threadIdx.x;
    if (i >= n) return;
    if (dens[i] > THRESH) {
        int ix = clamp127((int)(coords[3 * i + 0] * 127.0f));
        int iy = clamp127((int)(coords[3 * i + 1] * 127.0f));
        int iz = clamp127((int)(coords[3 * i + 2] * 127.0f));
        int lin = (ix << 14) | (iy << 7) | iz;
        updated[lin] = 1.0f;
        atomicOr(&gmask[lin >> 5], 1u << (lin & 31));
    }
}

// K3: stage full 256KB bitmask into LDS (async global->LDS on gfx1250),
// then 9 x ds_load_b128 per point cover all 27 clamped neighbor probes.
__global__ void og_gather(const float* __restrict__ coords,
                          const unsigned* __restrict__ gmask,
                          float* __restrict__ occupancy,
                          float* __restrict__ aoe, int n) {
    __shared__ unsigned smask[NWORDS];                    // 256 KB static LDS
    const int tid  = threadIdx.x;
    const int nthr = blockDim.x;

#if HAVE_ASYNC_LDS
    {
        const uint4* gsrc = (const uint4*)gmask;
        uint4*       ldst = (uint4*)smask;
        for (int i = tid; i < NCHUNK; i += nthr) {
            __builtin_amdgcn_global_load_async_to_lds_b128(
                (v4i*)(gsrc + i), (v4i*)(ldst + i), 0, 0);
        }
#if __has_builtin(__builtin_amdgcn_s_wait_asynccnt)
        __builtin_amdgcn_s_wait_asynccnt(0);
#else
        asm volatile("s_wait_asynccnt 0" ::: "memory");
#endif
    }
#else
    {
        const uint4* gsrc = (const uint4*)gmask;
        uint4*       ldst = (uint4*)smask;
        for (int i = tid; i < NCHUNK; i += nthr) ldst[i] = gsrc[i];
    }
#endif
    __syncthreads();

    const int total = gridDim.x * nthr;
    for (int i = blockIdx.x * nthr + tid; i < n; i += total) {
        int ix = clamp127((int)(coords[3 * i + 0] * 127.0f));
        int iy = clamp127((int)(coords[3 * i + 1] * 127.0f));
        int iz = clamp127((int)(coords[3 * i + 2] * 127.0f));

        int zm = iz - 1 < 0 ? 0 : iz - 1;
        int zp = iz + 1 > 127 ? 127 : iz + 1;
        int q0 = zm >> 5, b0 = zm & 31;
        int q1 = iz >> 5, b1 = iz & 31;
        int q2 = zp >> 5, b2 = zp & 31;

        unsigned cnt = 0u, occb = 0u;
#pragma unroll
        for (int dx = -1; dx <= 1; ++dx) {
            int xx = clamp127(ix + dx);
#pragma unroll
            for (int dy = -1; dy <= 1; ++dy) {
                int yy = clamp127(iy + dy);
                int row = (xx << 7) | yy;                 // 128-bit z-row, 16B aligned
                uint4 r = ((const uint4*)smask)[row];     // ds_load_b128
                unsigned w0 = q0 == 0 ? r.x : q0 == 1 ? r.y : q0 == 2 ? r.z : r.w;
                unsigned w1 = q1 == 0 ? r.x : q1 == 1 ? r.y : q1 == 2 ? r.z : r.w;
                unsigned w2 = q2 == 0 ? r.x : q2 == 1 ? r.y : q2 == 2 ? r.z : r.w;
                cnt += (w0 >> b0) & 1u;
                cnt += (w1 >> b1) & 1u;
                cnt += (w2 >> b2) & 1u;
                if (dx == 0 && dy == 0) occb = (w1 >> b1) & 1u;
            }
        }
        occupancy[i] = (float)occb;
        aoe[i]       = (float)cnt / 27.0f;
    }
}

extern "C" void kernel_launch(void* const* d_in, const int* in_sizes, int n_in,
                              void* d_out, int out_size, void* d_ws, size_t ws_size,
                              hipStream_t stream) {
    const float* coords = (const float*)d_in[0];   // (N,3) f32
    const float* dens   = (const float*)d_in[1];   // (N,)  f32
    const float* grid   = (const float*)d_in[2];   // 128^3 f32
    const int n = in_sizes[1];

    float* updated   = (float*)d_out;              // 128^3
    float* occupancy = updated + NCELLS;           // N
    float* aoe       = occupancy + n;              // N
    unsigned* gmask  = (unsigned*)d_ws;            // 256 KB bitmask

    og_pack_copy<<<NWORDS / 256, 256, 0, stream>>>(grid, updated, gmask);
    og_scatter<<<(n + 255) / 256, 256, 0, stream>>>(coords, dens, updated, gmask, n);
    og_gather<<<128, 1024, 0, stream>>>(coords, gmask, occupancy, aoe, n);
}